// Lift3DClipReplacementGraspNet_16733192585543
// MI455X (gfx1250) — compile-verified
//
#include <hip/hip_runtime.h>

#define B_   16
#define N_   20000
#define S_   2048
#define M_   1024
#define F_   768
#define OUT_ 256

typedef __attribute__((ext_vector_type(2))) float v2f;
typedef __attribute__((ext_vector_type(8))) float v8f;

// ---------------------------------------------------------------------------
// Kernel 1: nearest-center argmin per seed.
// The reference's per-batch centering+scaling is argmin-invariant (translation
// cancels in seed-ctr, positive scale is monotone), so point_cloud stats are
// skipped entirely. Centers for one batch (1024*3 f32 = 12KB) staged in LDS;
// every lane scans all 1024 centers (LDS broadcast, no bank conflicts).
// ---------------------------------------------------------------------------
__global__ __launch_bounds__(256) void nn_argmin_kernel(
    const float* __restrict__ seed_xyz,   // (B,S,3)
    const float* __restrict__ centers,    // (B,M,3)
    int* __restrict__ idx_out) {          // (B,S)
  __shared__ float cs[M_ * 3];
  const int b = blockIdx.y;
  const float* cb = centers + (size_t)b * M_ * 3;
  for (int i = threadIdx.x; i < M_ * 3; i += 256) cs[i] = cb[i];
  __syncthreads();

  const int s = blockIdx.x * 256 + threadIdx.x;
  const float* sp = seed_xyz + ((size_t)b * S_ + s) * 3;
  const float sx = sp[0], sy = sp[1], sz = sp[2];

  float best = 3.402823466e38f;
  int bi = 0;
#pragma unroll 4
  for (int m = 0; m < M_; ++m) {
    const float dx = sx - cs[m * 3 + 0];
    const float dy = sy - cs[m * 3 + 1];
    const float dz = sz - cs[m * 3 + 2];
    const float d = dx * dx + dy * dy + dz * dz;
    if (d < best) { best = d; bi = m; }   // strict < keeps first index (matches jnp.argmin)
  }
  idx_out[(size_t)b * S_ + s] = bi;
}

// ---------------------------------------------------------------------------
// Kernel 2: fused gather + fp32 WMMA GEMM + bias.
//   out[b, o, s] = sum_f W[o,f] * patch_tok[b, idx[b,s], f] + bias[o]
// Per wave: 32(OUT) x 32(S) register tile = 4 accumulators, K stepped by 4
// with V_WMMA_F32_16X16X4_F32. A-operand layout (16x4 f32): lane<16 holds
// {K,K+1} of row M=lane, lane>=16 holds {K+2,K+3} of row M=lane-16; B mirrors
// with N in place of M. C/D: lane<16 -> M=vgpr, lane>=16 -> M=vgpr+8, N=lane&15.
// Whole hot set (patch_tok 50MB + W 0.75MB) lives in the 192MB L2.
// ---------------------------------------------------------------------------
__global__ __launch_bounds__(256) void gather_gemm_wmma_kernel(
    const float* __restrict__ patch_tok,  // (B,M,F)
    const float* __restrict__ Wm,         // (OUT,F)
    const float* __restrict__ bias,       // (OUT)
    const int* __restrict__ idx,          // (B,S)
    float* __restrict__ out) {            // (B,OUT,S)
  const int lane = threadIdx.x & 31;
  const int wave = threadIdx.x >> 5;
  const int b    = blockIdx.z;
  const int row0 = blockIdx.y * 32;                    // OUT tile base (32 rows)
  const int s0   = blockIdx.x * 256 + wave * 32;       // S tile base (32 seeds)

  const int half = lane >> 4;   // selects K pair {0,1} vs {2,3}
  const int ln   = lane & 15;

  // A fragments: two row groups of W
  const float* a0p = Wm + (size_t)(row0 + ln) * F_ + 2 * half;
  const float* a1p = Wm + (size_t)(row0 + 16 + ln) * F_ + 2 * half;

  // B fragments: two gathered patch rows (one per 16-seed group)
  const int r0 = idx[(size_t)b * S_ + s0 + ln];
  const int r1 = idx[(size_t)b * S_ + s0 + 16 + ln];
  const float* b0p = patch_tok + ((size_t)b * M_ + r0) * F_ + 2 * half;
  const float* b1p = patch_tok + ((size_t)b * M_ + r1) * F_ + 2 * half;

  v8f acc00 = {}, acc01 = {}, acc10 = {}, acc11 = {};

#pragma unroll 4
  for (int k = 0; k < F_; k += 4) {
    const v2f a0 = *(const v2f*)(a0p + k);
    const v2f a1 = *(const v2f*)(a1p + k);
    const v2f b0 = *(const v2f*)(b0p + k);
    const v2f b1 = *(const v2f*)(b1p + k);
    acc00 = __builtin_amdgcn_wmma_f32_16x16x4_f32(false, a0, false, b0,
                                                  (short)0, acc00, false, false);
    acc01 = __builtin_amdgcn_wmma_f32_16x16x4_f32(false, a0, false, b1,
                                                  (short)0, acc01, false, false);
    acc10 = __builtin_amdgcn_wmma_f32_16x16x4_f32(false, a1, false, b0,
                                                  (short)0, acc10, false, false);
    acc11 = __builtin_amdgcn_wmma_f32_16x16x4_f32(false, a1, false, b1,
                                                  (short)0, acc11, false, false);
  }

  // Store: D element (vgpr v, lane) -> M = v + 8*half, N = ln. Coalesced in S.
  const int mb0 = row0 + 8 * half;        // rows for acc00/acc01
  const int mb1 = row0 + 16 + 8 * half;   // rows for acc10/acc11
  float* o00 = out + ((size_t)b * OUT_ + mb0) * S_ + s0 + ln;
  float* o10 = out + ((size_t)b * OUT_ + mb1) * S_ + s0 + ln;
#pragma unroll
  for (int v = 0; v < 8; ++v) {
    const float bv0 = bias[mb0 + v];
    const float bv1 = bias[mb1 + v];
    o00[(size_t)v * S_]      = acc00[v] + bv0;
    o00[(size_t)v * S_ + 16] = acc01[v] + bv0;
    o10[(size_t)v * S_]      = acc10[v] + bv1;
    o10[(size_t)v * S_ + 16] = acc11[v] + bv1;
  }
}

extern "C" void kernel_launch(void* const* d_in, const int* in_sizes, int n_in,
                              void* d_out, int out_size, void* d_ws, size_t ws_size,
                              hipStream_t stream) {
  (void)in_sizes; (void)n_in; (void)out_size; (void)ws_size;
  // setup_inputs order: 0=point_cloud(unused), 1=seed_xyz, 2=patch_tok,
  //                     3=centers, 4=W, 5=b
  const float* seed_xyz  = (const float*)d_in[1];
  const float* patch_tok = (const float*)d_in[2];
  const float* centers   = (const float*)d_in[3];
  const float* Wm        = (const float*)d_in[4];
  const float* bias      = (const float*)d_in[5];
  float* out = (float*)d_out;
  int* idx = (int*)d_ws;   // B*S ints = 128KB scratch

  dim3 g1(S_ / 256, B_, 1);                 // (8, 16)
  nn_argmin_kernel<<<g1, 256, 0, stream>>>(seed_xyz, centers, idx);

  dim3 g2(S_ / 256, OUT_ / 32, B_);         // (8, 8, 16) blocks, 8 waves each
  gather_gemm_wmma_kernel<<<g2, 256, 0, stream>>>(patch_tok, Wm, bias, idx, out);
}